// MaxUnpooling2D_88510686035968
// MI455X (gfx1250) — compile-verified
//
#include <hip/hip_runtime.h>

// MaxUnpooling2D scatter-add for MI455X (gfx1250).
// Shapes (from reference): updates/mask [B=32,H=64,W=64,C=128] fp32/int32,
// output [B=32,oH=128,oW=128,C=128] fp32, SAME padding, stride 2.
//
// All dims are powers of two:
//   per-batch input elems = 64*64*128 = 2^19
//   per-batch output elems = 128*128*128 = 2^21
//   mask m in [0, 2^21):  y = m>>14, x = (m>>7)&127  (channel bits of m ignored)
//   out_idx = (b<<21) | (m & 0x1FFF80) | c        (disjoint bit fields)

__device__ __forceinline__ void atomic_add_f32_dev(float* p, float v) {
    // Non-returning fp32 atomic add, device scope (coherent across WGPs at L2).
    // Emits the gfx1250 VMEM atomic directly; avoids any CAS-loop fallback.
    asm volatile("global_atomic_add_f32 %0, %1, off scope:SCOPE_DEV"
                 :
                 : "v"(p), "v"(v)
                 : "memory");
}

__global__ __launch_bounds__(256) void fill_zero_kernel(float4* __restrict__ out, int n4) {
    int i = blockIdx.x * blockDim.x + threadIdx.x;
    if (i < n4) {
        out[i] = make_float4(0.0f, 0.0f, 0.0f, 0.0f);  // global_store_b128
    }
}

__global__ __launch_bounds__(256) void unpool_scatter_kernel(
        const float4* __restrict__ updates4,
        const int4*   __restrict__ mask4,
        float*        __restrict__ out,
        int n4) {
    int i = blockIdx.x * blockDim.x + threadIdx.x;
    if (i >= n4) return;

    const float4 u = updates4[i];   // global_load_b128, fully coalesced
    const int4   m = mask4[i];      // global_load_b128, fully coalesced

    const unsigned base  = ((unsigned)i) << 2;          // flat element index
    const unsigned bterm = (base >> 19) << 21;          // batch field
    const unsigned c     = base & 127u;                 // channel (multiple of 4)

    // Per-element output indices: batch | (y,x from mask) | positional channel.
    const unsigned o0 = bterm | ((unsigned)m.x & 0x1FFF80u) | (c + 0u);
    const unsigned o1 = bterm | ((unsigned)m.y & 0x1FFF80u) | (c + 1u);
    const unsigned o2 = bterm | ((unsigned)m.z & 0x1FFF80u) | (c + 2u);
    const unsigned o3 = bterm | ((unsigned)m.w & 0x1FFF80u) | (c + 3u);

    // Duplicate mask values must sum (tf.scatter_nd semantics) -> atomics.
    atomic_add_f32_dev(out + o0, u.x);
    atomic_add_f32_dev(out + o1, u.y);
    atomic_add_f32_dev(out + o2, u.z);
    atomic_add_f32_dev(out + o3, u.w);
}

extern "C" void kernel_launch(void* const* d_in, const int* in_sizes, int n_in,
                              void* d_out, int out_size, void* d_ws, size_t ws_size,
                              hipStream_t stream) {
    const float* updates = (const float*)d_in[0];
    const int*   mask    = (const int*)d_in[1];
    float*       out     = (float*)d_out;

    // 1) Zero the output: 64M floats = 16M float4 stores (256 MiB @ 23.3 TB/s ~ 11 us).
    {
        const int n4 = out_size >> 2;                 // 16,777,216
        const int threads = 256;
        const int blocks = (n4 + threads - 1) / threads;
        fill_zero_kernel<<<blocks, threads, 0, stream>>>((float4*)out, n4);
    }

    // 2) Scatter-add: 16M elements, 4 per thread (128-bit loads + 4 fp32 atomics).
    {
        const int n_elems = in_sizes[0];              // 16,777,216
        const int n4 = n_elems >> 2;                  // 4,194,304 threads
        const int threads = 256;
        const int blocks = (n4 + threads - 1) / threads;
        unpool_scatter_kernel<<<blocks, threads, 0, stream>>>(
            (const float4*)updates, (const int4*)mask, out, n4);
    }
}